// Encoder_38809324486723
// MI455X (gfx1250) — compile-verified
//
#include <hip/hip_runtime.h>
#include <hip/hip_bf16.h>

#define Hh 4
#define Cc 32
#define HID 128
#define LAT 32
#define NEG_SLOPE 0.2f

typedef __attribute__((ext_vector_type(2))) float v2f;
typedef __attribute__((ext_vector_type(8))) float v8f;

// ---------- utility: float atomic max via sign-split int trick ----------
__device__ __forceinline__ void atomicMaxFloat(float* addr, float val) {
    if (val >= 0.0f) {
        atomicMax((int*)addr, __float_as_int(val));
    } else {
        atomicMin((unsigned int*)addr, __float_as_uint(val));
    }
}

// ---------- K0: zero the accumulated regions (deg + ea_sum, contiguous) ----------
__global__ void k_zero(float* p, long long n) {
    long long i = (long long)blockIdx.x * blockDim.x + threadIdx.x;
    if (i < n) p[i] = 0.0f;
}

// ---------- Kw: waw[j*4+h] = sum_c W_edge[j*128 + h*32 + c] * att_edge[h*32 + c] ----------
__global__ void k_waw(const float* __restrict__ We, const float* __restrict__ att_edge,
                      float* __restrict__ waw) {
    int t = threadIdx.x;
    if (t < 28) {
        int j = t >> 2, h = t & 3;
        float s = 0.0f;
        for (int c = 0; c < Cc; ++c)
            s += We[j * HID + h * Cc + c] * att_edge[h * Cc + c];
        waw[j * 4 + h] = s;
    }
}

// ---------- K2: degree + edge_attr segment sums over dst ----------
__global__ void k_deg_ea(const long long* __restrict__ dst, const float* __restrict__ ea,
                         float* __restrict__ deg, float* __restrict__ ea7, int E) {
    int e = blockIdx.x * blockDim.x + threadIdx.x;
    if (e >= E) return;
    int d = (int)dst[e];
    atomicAdd(deg + d, 1.0f);
    #pragma unroll
    for (int j = 0; j < 7; ++j)
        atomicAdd(ea7 + (long long)d * 7 + j, ea[(long long)e * 7 + j]);
}

// ---------- K3: per-node xp = x @ W_gat, a_src/a_dst, self-loop alpha (inits m) ----------
__global__ void k_node(const float* __restrict__ x, const float* __restrict__ Wg,
                       const float* __restrict__ att_src, const float* __restrict__ att_dst,
                       const float* __restrict__ deg, const float* __restrict__ ea7,
                       const float* __restrict__ waw,
                       float* __restrict__ xp, float* __restrict__ asrc,
                       float* __restrict__ adst, float* __restrict__ aself,
                       float* __restrict__ m, int N) {
    int n = blockIdx.x * blockDim.x + threadIdx.x;
    if (n >= N) return;
    float x0 = x[n * 3 + 0], x1 = x[n * 3 + 1], x2 = x[n * 3 + 2];
    float dd = fmaxf(deg[n], 1.0f);
    float mean[7];
    #pragma unroll
    for (int j = 0; j < 7; ++j) mean[j] = ea7[(long long)n * 7 + j] / dd;

    float asr[Hh] = {0,0,0,0}, ads[Hh] = {0,0,0,0};
    for (int k = 0; k < HID; ++k) {
        float v = x0 * Wg[k] + x1 * Wg[HID + k] + x2 * Wg[2 * HID + k];
        xp[(long long)n * HID + k] = v;
        int h = k >> 5;
        asr[h] += v * att_src[k];
        ads[h] += v * att_dst[k];
    }
    #pragma unroll
    for (int h = 0; h < Hh; ++h) {
        float ae = 0.0f;
        #pragma unroll
        for (int j = 0; j < 7; ++j) ae += mean[j] * waw[j * 4 + h];
        float s = asr[h] + ads[h] + ae;
        s = s > 0.0f ? s : NEG_SLOPE * s;   // leaky_relu
        asrc[n * 4 + h]  = asr[h];
        adst[n * 4 + h]  = ads[h];
        aself[n * 4 + h] = s;
        m[n * 4 + h]     = s;               // self-loop seeds the segment max
    }
}

// ---------- K4: per real edge raw alpha + segment max ----------
__global__ void k_edge_alpha(const long long* __restrict__ src, const long long* __restrict__ dst,
                             const float* __restrict__ ea, const float* __restrict__ waw,
                             const float* __restrict__ asrc, const float* __restrict__ adst,
                             float* __restrict__ araw, float* __restrict__ m, int E) {
    int e = blockIdx.x * blockDim.x + threadIdx.x;
    if (e >= E) return;
    int s = (int)src[e], d = (int)dst[e];
    float eav[7];
    #pragma unroll
    for (int j = 0; j < 7; ++j) eav[j] = ea[(long long)e * 7 + j];
    #pragma unroll
    for (int h = 0; h < Hh; ++h) {
        float ae = 0.0f;
        #pragma unroll
        for (int j = 0; j < 7; ++j) ae += eav[j] * waw[j * 4 + h];
        float r = asrc[s * 4 + h] + adst[d * 4 + h] + ae;
        r = r > 0.0f ? r : NEG_SLOPE * r;
        araw[(long long)e * 4 + h] = r;
        atomicMaxFloat(m + d * 4 + h, r);
    }
}

// ---------- K5: seed denom + acc with self-loop message ----------
__global__ void k_self_init(const float* __restrict__ aself, const float* __restrict__ m,
                            const float* __restrict__ xp,
                            float* __restrict__ denom, float* __restrict__ acc, int N) {
    int n = blockIdx.x * blockDim.x + threadIdx.x;
    if (n >= N) return;
    #pragma unroll
    for (int h = 0; h < Hh; ++h) {
        float es = expf(aself[n * 4 + h] - m[n * 4 + h]);
        denom[n * 4 + h] = es;
        for (int c = 0; c < Cc; ++c)
            acc[(long long)n * HID + h * Cc + c] = es * xp[(long long)n * HID + h * Cc + c];
    }
}

// ---------- K6: wave-per-edge message scatter (lane = channel) ----------
__global__ void k_msg(const long long* __restrict__ src, const long long* __restrict__ dst,
                      const float* __restrict__ araw, const float* __restrict__ m,
                      const float* __restrict__ xp,
                      float* __restrict__ denom, float* __restrict__ acc, int E) {
    int e = blockIdx.x * 8 + (threadIdx.x >> 5);
    if (e >= E) return;
    int lane = threadIdx.x & 31;
    int s = (int)src[e], d = (int)dst[e];
    float w[Hh];
    #pragma unroll
    for (int h = 0; h < Hh; ++h)
        w[h] = expf(araw[(long long)e * 4 + h] - m[d * 4 + h]);
    #pragma unroll
    for (int h = 0; h < Hh; ++h)
        atomicAdd(acc + (long long)d * HID + h * Cc + lane,
                  w[h] * xp[(long long)s * HID + h * Cc + lane]);
    if (lane < Hh) atomicAdd(denom + d * 4 + lane, w[lane]);
}

// ---------- K7: normalize + bias + ELU (in place: acc becomes h) ----------
__global__ void k_elu(float* __restrict__ acc, const float* __restrict__ denom,
                      const float* __restrict__ bias, int N) {
    int n = blockIdx.x * blockDim.x + threadIdx.x;
    if (n >= N) return;
    for (int k = 0; k < HID; ++k) {
        int h = k >> 5;
        float v = acc[(long long)n * HID + k] / denom[n * 4 + h] + bias[k];
        acc[(long long)n * HID + k] = v > 0.0f ? v : expm1f(v);
    }
}

// ---------- K8: layer1 h @ W1 + b1, PReLU — WMMA f32 16x16x4, 8 waves = 8 col tiles ----------
__global__ void k_mlp1(const float* __restrict__ hf, const float* __restrict__ W1,
                       const float* __restrict__ b1, const float* __restrict__ prelu_a,
                       float* __restrict__ h1, int N) {
    const int lane  = threadIdx.x & 31;
    const int ntile = threadIdx.x >> 5;          // 0..7
    const int row0  = blockIdx.x * 16;
    const int col0  = ntile * 16;
    const int rload = min(row0 + (lane & 15), N - 1);
    const int col   = col0 + (lane & 15);
    const int khalf = (lane >> 4) * 2;           // lanes 16-31 take K+2,K+3
    v8f acc = {};
    #pragma unroll 8
    for (int k0 = 0; k0 < HID; k0 += 4) {
        v2f a, b;
        a.x = hf[(long long)rload * HID + k0 + khalf];
        a.y = hf[(long long)rload * HID + k0 + khalf + 1];
        b.x = W1[(long long)(k0 + khalf) * HID + col];
        b.y = W1[(long long)(k0 + khalf + 1) * HID + col];
        acc = __builtin_amdgcn_wmma_f32_16x16x4_f32(false, a, false, b,
                                                    (short)0, acc, false, false);
    }
    float aL = *prelu_a;
    float bb = b1[col];
    #pragma unroll
    for (int v = 0; v < 8; ++v) {
        int rr = row0 + v + 8 * (lane >> 4);
        if (rr < N) {
            float val = acc[v] + bb;
            val = val > 0.0f ? val : aL * val;   // PReLU
            h1[(long long)rr * HID + col] = val;
        }
    }
}

// ---------- K9: layer2 h1 @ W2 + b2 — WMMA f32 16x16x4, 2 waves = 2 col tiles ----------
__global__ void k_mlp2(const float* __restrict__ h1, const float* __restrict__ W2,
                       const float* __restrict__ b2, float* __restrict__ out, int N) {
    const int lane  = threadIdx.x & 31;
    const int ntile = threadIdx.x >> 5;          // 0..1
    const int row0  = blockIdx.x * 16;
    const int col0  = ntile * 16;
    const int rload = min(row0 + (lane & 15), N - 1);
    const int col   = col0 + (lane & 15);
    const int khalf = (lane >> 4) * 2;
    v8f acc = {};
    #pragma unroll 8
    for (int k0 = 0; k0 < HID; k0 += 4) {
        v2f a, b;
        a.x = h1[(long long)rload * HID + k0 + khalf];
        a.y = h1[(long long)rload * HID + k0 + khalf + 1];
        b.x = W2[(long long)(k0 + khalf) * LAT + col];
        b.y = W2[(long long)(k0 + khalf + 1) * LAT + col];
        acc = __builtin_amdgcn_wmma_f32_16x16x4_f32(false, a, false, b,
                                                    (short)0, acc, false, false);
    }
    float bb = b2[col];
    #pragma unroll
    for (int v = 0; v < 8; ++v) {
        int rr = row0 + v + 8 * (lane >> 4);
        if (rr < N)
            out[(long long)rr * LAT + col] = acc[v] + bb;
    }
}

extern "C" void kernel_launch(void* const* d_in, const int* in_sizes, int n_in,
                              void* d_out, int out_size, void* d_ws, size_t ws_size,
                              hipStream_t stream) {
    const float*     x        = (const float*)d_in[0];
    const long long* ei       = (const long long*)d_in[1];
    const float*     ea       = (const float*)d_in[2];
    const float*     Wg       = (const float*)d_in[3];
    const float*     att_src  = (const float*)d_in[4];
    const float*     att_dst  = (const float*)d_in[5];
    const float*     We       = (const float*)d_in[6];
    const float*     att_edge = (const float*)d_in[7];
    const float*     bias     = (const float*)d_in[8];
    const float*     W1       = (const float*)d_in[9];
    const float*     b1       = (const float*)d_in[10];
    const float*     pa       = (const float*)d_in[11];
    const float*     W2       = (const float*)d_in[12];
    const float*     b2       = (const float*)d_in[13];
    float*           out      = (float*)d_out;

    const int N = in_sizes[0] / 3;
    const int E = in_sizes[2] / 7;
    const long long* src = ei;
    const long long* dst = ei + E;

    // workspace layout (floats)
    float* ws    = (float*)d_ws;
    float* deg   = ws;                 ws += N;            // zeroed
    float* ea7   = ws;                 ws += 7LL * N;      // zeroed (contiguous w/ deg)
    float* asrc  = ws;                 ws += 4LL * N;
    float* adst  = ws;                 ws += 4LL * N;
    float* aself = ws;                 ws += 4LL * N;
    float* m     = ws;                 ws += 4LL * N;
    float* denom = ws;                 ws += 4LL * N;
    float* xp    = ws;                 ws += (long long)HID * N;
    float* acc   = ws;                 ws += (long long)HID * N;  // reused as h after ELU
    float* h1    = ws;                 ws += (long long)HID * N;
    float* waw   = ws;                 ws += 32;
    float* araw  = ws;                 ws += 4LL * E;

    const int T = 256;
    long long zn = 8LL * N;  // deg + ea7

    k_zero<<<(int)((zn + T - 1) / T), T, 0, stream>>>(deg, zn);
    k_waw<<<1, 32, 0, stream>>>(We, att_edge, waw);
    k_deg_ea<<<(E + T - 1) / T, T, 0, stream>>>(dst, ea, deg, ea7, E);
    k_node<<<(N + T - 1) / T, T, 0, stream>>>(x, Wg, att_src, att_dst, deg, ea7, waw,
                                              xp, asrc, adst, aself, m, N);
    k_edge_alpha<<<(E + T - 1) / T, T, 0, stream>>>(src, dst, ea, waw, asrc, adst, araw, m, E);
    k_self_init<<<(N + T - 1) / T, T, 0, stream>>>(aself, m, xp, denom, acc, N);
    k_msg<<<(E + 7) / 8, 256, 0, stream>>>(src, dst, araw, m, xp, denom, acc, E);
    k_elu<<<(N + T - 1) / T, T, 0, stream>>>(acc, denom, bias, N);

    int mtiles = (N + 15) / 16;
    k_mlp1<<<mtiles, 256, 0, stream>>>(acc, W1, b1, pa, h1, N);   // 8 waves = 8 col tiles
    k_mlp2<<<mtiles, 64, 0, stream>>>(h1, W2, b2, out, N);        // 2 waves = 2 col tiles
}